// ClassifyingReconstructionLoss_34385508171852
// MI455X (gfx1250) — compile-verified
//
#include <hip/hip_runtime.h>
#include <hip/hip_bf16.h>

// Problem dims (fixed by the reference)
#define NSTEPS 8
#define NBATCH 128
#define NVOCAB 32000
#define NROWS  (NSTEPS * NBATCH)   // 1024 independent softmax rows
#define TILE   4096                // floats per LDS tile (16 KB)
#define NT     8                   // ceil(32000 / 4096)
#define BLK    256                 // 8 waves of 32
#define CHUNK  (TILE / (BLK * 4))  // 4 async b128 per thread per tile

typedef __attribute__((ext_vector_type(2))) float v2f;
typedef __attribute__((ext_vector_type(8))) float v8f;

// ---- CDNA5 async global->LDS copy (ASYNCcnt-tracked), inline asm form ----
__device__ __forceinline__ void async_b128(unsigned lds_byte, int glb_byte_off,
                                           const void* sbase) {
  // GVS mode: mem_addr = SADDR(sgpr pair) + VADDR(32-bit byte offset)
  asm volatile("global_load_async_to_lds_b128 %0, %1, %2"
               :
               : "v"(lds_byte), "v"(glb_byte_off), "s"(sbase)
               : "memory");
}

__device__ __forceinline__ void wait_async_le4() {
  asm volatile("s_wait_asynccnt 4" ::: "memory");
}
__device__ __forceinline__ void wait_async_0() {
  asm volatile("s_wait_asynccnt 0" ::: "memory");
}

// Online log-sum-exp accumulator update (one exp per element)
__device__ __forceinline__ void online(float x, float& m, float& s) {
  if (x <= m) {
    s += __expf(x - m);
  } else {
    s = s * __expf(m - x) + 1.0f;
    m = x;
  }
}

// Combine two (max, sumexp) partials
__device__ __forceinline__ void combine(float& m, float& s, float m2, float s2) {
  float M = fmaxf(m, m2);
  s = s * __expf(m - M) + s2 * __expf(m2 - M);
  m = M;
}

__global__ __launch_bounds__(BLK)
void rowlse_kernel(const float* __restrict__ p, const float* __restrict__ ypred,
                   const int* __restrict__ ytrue, float* __restrict__ rowloss) {
  __shared__ __align__(16) float tile[2 * TILE];
  const int tid = threadIdx.x;
  const int row = blockIdx.x;                      // row = n*128 + b
  const float* rowp = ypred + (size_t)row * NVOCAB;

  auto prefetch = [&](int t) {
    const int buf = t & 1;
#pragma unroll
    for (int c = 0; c < CHUNK; ++c) {
      int idx  = t * TILE + c * (BLK * 4) + tid * 4;
      int gidx = idx < NVOCAB ? idx : (NVOCAB - 4);  // clamp: uniform issue count
      // generic shared pointer: low 32 bits == LDS byte offset (LDS aperture)
      unsigned lds = (unsigned)(size_t)&tile[buf * TILE + c * (BLK * 4) + tid * 4];
      async_b128(lds, gidx * 4, (const void*)rowp);
    }
  };

  prefetch(0);
  float m = -3.402823466e+38f;  // -FLT_MAX
  float s = 0.0f;

  for (int t = 0; t < NT; ++t) {
    if (t + 1 < NT) {
      prefetch(t + 1);    // double-buffer: overlaps with consumption of tile t
      wait_async_le4();   // only tile t+1's 4 groups may remain outstanding
    } else {
      wait_async_0();
    }
    __syncthreads();      // all waves' tile-t data is in LDS
    const int buf = t & 1;
#pragma unroll
    for (int c = 0; c < CHUNK; ++c) {
      int idx = t * TILE + c * (BLK * 4) + tid * 4;
      if (idx < NVOCAB) {
        const float4 v = *reinterpret_cast<const float4*>(
            &tile[buf * TILE + c * (BLK * 4) + tid * 4]);
        online(v.x, m, s); online(v.y, m, s);
        online(v.z, m, s); online(v.w, m, s);
      }
    }
    __syncthreads();      // safe to overwrite this buffer two tiles later
  }

  // wave32 reduction of (m, s)
#pragma unroll
  for (int off = 16; off > 0; off >>= 1) {
    float m2 = __shfl_xor(m, off, 32);
    float s2 = __shfl_xor(s, off, 32);
    combine(m, s, m2, s2);
  }
  __shared__ float wm[BLK / 32], wsum[BLK / 32];
  if ((tid & 31) == 0) { wm[tid >> 5] = m; wsum[tid >> 5] = s; }
  __syncthreads();
  if (tid == 0) {
    float M = wm[0], S = wsum[0];
#pragma unroll
    for (int w = 1; w < BLK / 32; ++w) combine(M, S, wm[w], wsum[w]);
    const int b   = row & (NBATCH - 1);
    const int col = ytrue[b] - 1;
    const float xt  = rowp[col];          // served from L2 (whole tensor fits)
    const float lse = M + logf(S);
    rowloss[row] = p[row] * (lse - xt);   // p is (8,128) row-major -> p[row]
  }
}

// Deterministic final reduction of 1024 row losses; wave sum via exact
// f32 WMMA ones-trick (D[m,n] = sum_k B[k,n] with A == ones).
__global__ __launch_bounds__(256)
void final_reduce_kernel(const float* __restrict__ rowloss, float* __restrict__ out) {
  const int tid = threadIdx.x;
  float v = rowloss[tid] + rowloss[tid + 256] + rowloss[tid + 512] + rowloss[tid + 768];
#if __has_builtin(__builtin_amdgcn_wmma_f32_16x16x4_f32)
  // EXEC is all ones here: full 256-thread block, no divergence.
  v2f a; a[0] = 1.0f; a[1] = 1.0f;   // A (16x4) all ones
  v2f b; b[0] = v;    b[1] = 0.0f;   // B (4x16): column n holds v[n], v[n+16]
  v8f c = {};
  v8f d = __builtin_amdgcn_wmma_f32_16x16x4_f32(
      /*neg_a=*/false, a, /*neg_b=*/false, b,
      /*c_mod=*/(short)0, c, /*reuse_a=*/false, /*reuse_b=*/false);
  // d[0] in every lane l == v[l%16] + v[l%16 + 16]; finish 16-way with shuffles
  float r = d[0];
#pragma unroll
  for (int off = 8; off > 0; off >>= 1) r += __shfl_xor(r, off, 32);
#else
  float r = v;
#pragma unroll
  for (int off = 16; off > 0; off >>= 1) r += __shfl_xor(r, off, 32);
#endif
  __shared__ float ws[8];
  if ((tid & 31) == 0) ws[tid >> 5] = r;
  __syncthreads();
  if (tid == 0) {
    float t = 0.0f;
#pragma unroll
    for (int w = 0; w < 8; ++w) t += ws[w];
    out[0] = t * (1.0f / NBATCH);
  }
}

extern "C" void kernel_launch(void* const* d_in, const int* in_sizes, int n_in,
                              void* d_out, int out_size, void* d_ws, size_t ws_size,
                              hipStream_t stream) {
  const float* p     = (const float*)d_in[0];  // (8,128) f32
  const float* ypred = (const float*)d_in[1];  // (8,128,32000) f32
  const int*   ytrue = (const int*)d_in[2];    // (128,) int
  // d_in[3] = pad_id (unused by the math)
  float* rowloss = (float*)d_ws;               // 1024 f32 partials

  rowlse_kernel<<<NROWS, BLK, 0, stream>>>(p, ypred, ytrue, rowloss);
  final_reduce_kernel<<<1, 256, 0, stream>>>(rowloss, (float*)d_out);
}